// WaveSplitBase_63067299774783
// MI455X (gfx1250) — compile-verified
//
#include <hip/hip_runtime.h>

typedef float v2f __attribute__((ext_vector_type(2)));
typedef float v8f __attribute__((ext_vector_type(8)));

#define Bn   4
#define Dn   512
#define Tn   16000
#define ITERS 100
#define FPB  128                 // frames per block
#define BPB  (Tn / FPB)          // 125 blocks per batch
#define NTHR 128                 // 4 wave32s

// workspace layout (floats)
#define OFF_M   0
#define SZ_M    (Bn * Tn * 4)            // 256000  cumulative 2x2 coeffs per frame
#define OFF_P   (OFF_M + SZ_M)
#define SZ_P    (Bn * BPB * 2 * Dn)      // 512000  per-block centroid partials
#define OFF_C   (OFF_P + SZ_P)
#define SZ_C    (Bn * 2 * Dn)            // 4096    centroids
#define OFF_C2  (OFF_C + SZ_C)           // 8       ||cent||^2

// One fused pass per iteration:
//  phase 1 (WMMA): dots of original v0,v1 vs both centroids for 16-frame tiles,
//                  per-frame argmin, update cumulative coeff matrix M
//  phase 2:        accumulate next-iteration centroid partials with new M
__global__ __launch_bounds__(NTHR)
void kmeans_step(const float* __restrict__ v, float* __restrict__ Mbuf,
                 float* __restrict__ partial, const float* __restrict__ cent,
                 const float* __restrict__ c2g, int first)
{
    __shared__ float  sCent[2 * Dn];
    __shared__ float  sC2[2];
    __shared__ float4 sMn[FPB];
    __shared__ float  sP[4][2][32][8];

    const int tid  = threadIdx.x;
    const int w    = tid >> 5;
    const int lane = tid & 31;
    const int blk  = blockIdx.x;          // 0..124
    const int b    = blockIdx.y;          // 0..3
    const int t0   = blk * FPB;
    const float* vb0 = v + (size_t)(b * 2 + 0) * Dn * Tn;
    const float* vb1 = v + (size_t)(b * 2 + 1) * Dn * Tn;

    if (!first) {
        for (int i = tid; i < 2 * Dn; i += NTHR) sCent[i] = cent[b * 2 * Dn + i];
        if (tid < 2) sC2[tid] = c2g[b * 2 + tid];
        __syncthreads();

        const int m  = lane & 15;         // frame within tile (A-matrix M index)
        const int kh = lane >> 4;         // lane-half -> K offset 0 / 2
        for (int tt = 0; tt < 2; ++tt) {  // 2 tiles of 16 frames per wave
            const int lf = w * 32 + tt * 16;
            const int t  = t0 + lf;
            v8f P0 = {}, P1 = {};
            const float* a0 = vb0 + (size_t)(kh * 2) * Tn + t + m;
            const float* a1 = vb1 + (size_t)(kh * 2) * Tn + t + m;
            const float2* bp = (const float2*)&sCent[(lane & 1) * Dn + kh * 2];
            for (int ch = 0; ch < Dn / 4; ++ch) {   // K = 512 in steps of 4
                v2f A0, A1, Bf;
                A0.x = a0[0]; A0.y = a0[Tn];
                A1.x = a1[0]; A1.y = a1[Tn];
                float2 bv = bp[0];
                Bf.x = bv.x;  Bf.y = bv.y;          // B[k][n] = cent[n][d0+k] (n>=2 unused)
                P0 = __builtin_amdgcn_wmma_f32_16x16x4_f32(false, A0, false, Bf, (short)0, P0, false, false);
                P1 = __builtin_amdgcn_wmma_f32_16x16x4_f32(false, A1, false, Bf, (short)0, P1, false, false);
                a0 += (size_t)4 * Tn; a1 += (size_t)4 * Tn; bp += 2;
            }
            // stage C/D fragments so each frame-lane can gather its 4 dots
            #pragma unroll
            for (int j = 0; j < 8; ++j) { sP[w][0][lane][j] = P0[j]; sP[w][1][lane][j] = P1[j]; }
            __syncthreads();
            if (lane < 16) {
                const int hl = 16 * (lane >> 3), jj = lane & 7;   // (m,n) -> lane n+16*(m>>3), vgpr m&7
                float p00 = sP[w][0][0 + hl][jj];
                float p01 = sP[w][0][1 + hl][jj];
                float p10 = sP[w][1][0 + hl][jj];
                float p11 = sP[w][1][1 + hl][jj];
                const int tf = t + lane;
                float4 Mo = *(const float4*)&Mbuf[((size_t)b * Tn + tf) * 4];
                float dot00 = Mo.x * p00 + Mo.y * p10;
                float dot01 = Mo.x * p01 + Mo.y * p11;
                float dot10 = Mo.z * p00 + Mo.w * p10;
                float dot11 = Mo.z * p01 + Mo.w * p11;
                float s00 = sC2[0] - 2.f * dot00, s01 = sC2[1] - 2.f * dot01;
                float s10 = sC2[0] - 2.f * dot10, s11 = sC2[1] - 2.f * dot11;
                int i0 = s01 < s00;   // argmin (ties -> 0, matches jnp.argmin)
                int i1 = s11 < s10;
                float4 Mn;
                Mn.x = (i0 == 0 ? Mo.x : 0.f) + (i1 == 0 ? Mo.z : 0.f);
                Mn.y = (i0 == 0 ? Mo.y : 0.f) + (i1 == 0 ? Mo.w : 0.f);
                Mn.z = (i0 == 1 ? Mo.x : 0.f) + (i1 == 1 ? Mo.z : 0.f);
                Mn.w = (i0 == 1 ? Mo.y : 0.f) + (i1 == 1 ? Mo.w : 0.f);
                *(float4*)&Mbuf[((size_t)b * Tn + tf) * 4] = Mn;
                sMn[lf + lane] = Mn;
            }
            __syncthreads();
        }
    } else {
        const int tf = t0 + tid;
        *(float4*)&Mbuf[((size_t)b * Tn + tf) * 4] = make_float4(1.f, 0.f, 0.f, 1.f);
    }

    // ---- phase 2: accumulate centroid partials with NEW coefficients ----
    float c00[4], c01[4], c10[4], c11[4];
    if (!first) {
        #pragma unroll
        for (int g = 0; g < 4; ++g) {
            float4 mm = sMn[g * 32 + lane];
            c00[g] = mm.x; c01[g] = mm.y; c10[g] = mm.z; c11[g] = mm.w;
        }
    } else {
        #pragma unroll
        for (int g = 0; g < 4; ++g) { c00[g] = 1.f; c01[g] = 0.f; c10[g] = 0.f; c11[g] = 1.f; }
    }
    for (int d32 = 0; d32 < 4; ++d32) {     // each wave owns 128 d-values
        float st0 = 0.f, st1 = 0.f;
        for (int l = 0; l < 32; ++l) {
            const int d = w * 128 + d32 * 32 + l;
            const float* p0p = vb0 + (size_t)d * Tn + t0 + lane;
            const float* p1p = vb1 + (size_t)d * Tn + t0 + lane;
            float q0 = 0.f, q1 = 0.f;
            #pragma unroll
            for (int g = 0; g < 4; ++g) {
                float x0 = p0p[g * 32], x1 = p1p[g * 32];
                q0 += c00[g] * x0 + c01[g] * x1;
                q1 += c10[g] * x0 + c11[g] * x1;
            }
            #pragma unroll
            for (int mk = 1; mk < 32; mk <<= 1) {
                q0 += __shfl_xor(q0, mk, 32);
                q1 += __shfl_xor(q1, mk, 32);
            }
            if (lane == l) { st0 = q0; st1 = q1; }
        }
        const int d = w * 128 + d32 * 32 + lane;
        const size_t pb = (size_t)(b * BPB + blk) * 2 * Dn;
        partial[pb + d]      = st0;
        partial[pb + Dn + d] = st1;
    }
}

// deterministic partial -> centroid reduction, plus ||cent||^2
__global__ __launch_bounds__(256)
void kmeans_reduce(const float* __restrict__ partial, float* __restrict__ cent,
                   float* __restrict__ c2g)
{
    __shared__ float sred[256][2];
    const int b = blockIdx.x, tid = threadIdx.x;
    float a0 = 0.f, a1 = 0.f;
    #pragma unroll
    for (int k = 0; k < 4; ++k) {
        const int slot = tid + k * 256;     // slot = c*512 + d
        float s = 0.f;
        for (int blk = 0; blk < BPB; ++blk)
            s += partial[(size_t)(b * BPB + blk) * 2 * Dn + slot];
        const float c = s * (1.f / Tn);
        cent[(size_t)b * 2 * Dn + slot] = c;
        if (k < 2) a0 += c * c; else a1 += c * c;
    }
    sred[tid][0] = a0; sred[tid][1] = a1;
    __syncthreads();
    if (tid == 0) {
        float t0 = 0.f, t1 = 0.f;
        for (int i = 0; i < 256; ++i) { t0 += sred[i][0]; t1 += sred[i][1]; }
        c2g[b * 2 + 0] = t0; c2g[b * 2 + 1] = t1;
    }
}

// out[b][c][d][t] = M[c,0]*v0 + M[c,1]*v1
__global__ __launch_bounds__(256)
void kmeans_emit(const float* __restrict__ v, const float* __restrict__ Mbuf,
                 float* __restrict__ out)
{
    const int t = blockIdx.x * 256 + threadIdx.x;
    const int d = blockIdx.y, b = blockIdx.z;
    if (t >= Tn) return;
    float4 Mo = *(const float4*)&Mbuf[((size_t)b * Tn + t) * 4];
    const size_t base = (size_t)(b * 2) * Dn * Tn + (size_t)d * Tn + t;
    float x0 = v[base], x1 = v[base + (size_t)Dn * Tn];
    out[base]                      = Mo.x * x0 + Mo.y * x1;
    out[base + (size_t)Dn * Tn]    = Mo.z * x0 + Mo.w * x1;
}

extern "C" void kernel_launch(void* const* d_in, const int* in_sizes, int n_in,
                              void* d_out, int out_size, void* d_ws, size_t ws_size,
                              hipStream_t stream)
{
    const float* v = (const float*)d_in[0];
    float* ws      = (float*)d_ws;
    float* Mbuf    = ws + OFF_M;
    float* partial = ws + OFF_P;
    float* cent    = ws + OFF_C;
    float* c2v     = ws + OFF_C2;

    dim3 gA(BPB, Bn);
    kmeans_step<<<gA, NTHR, 0, stream>>>(v, Mbuf, partial, cent, c2v, 1);
    kmeans_reduce<<<Bn, 256, 0, stream>>>(partial, cent, c2v);
    for (int it = 0; it < ITERS; ++it) {
        kmeans_step<<<gA, NTHR, 0, stream>>>(v, Mbuf, partial, cent, c2v, 0);
        if (it + 1 < ITERS)
            kmeans_reduce<<<Bn, 256, 0, stream>>>(partial, cent, c2v);
    }
    kmeans_emit<<<dim3((Tn + 255) / 256, Dn, Bn), 256, 0, stream>>>(v, Mbuf, (float*)d_out);
}